// AGCNBlock_46299747450927
// MI455X (gfx1250) — compile-verified
//
#include <hip/hip_runtime.h>
#include <hip/hip_bf16.h>

// ---------------- problem constants (match reference) ----------------
#define BATCH   16
#define NNODES  2048
#define DDIM    128
#define HDIM    128
#define KMAXV   512
#define EPSV    1e-10f
#define NEGV    1e10f
#define PERCENTV 0.25f

typedef __attribute__((ext_vector_type(2))) float v2f;
typedef __attribute__((ext_vector_type(8))) float v8f;

// =====================================================================
// Tiled fp32 GEMM using V_WMMA_F32_16X16X4_F32.
// C[b] = op(A[b] @ B[b]) ; 128x128 block tile, BK=32, 4 waves/block
// (128 threads), each wave computes a 64x64 sub-tile as a 4x4 grid of
// 16x16 WMMA tiles -> 16 WMMAs per 8 LDS fragment loads per k-step.
// Template flags: RELU, BIAS (bias[n]), BT (B given transposed [Nn x K]),
// BSCALE (scale B row k by bscale[k] -> folds att*h), ROWSCALE (scale C row).
// =====================================================================
template<bool RELU, bool BIAS, bool BT, bool BSCALE, bool ROWSCALE>
__global__ __launch_bounds__(128)
void gemm_f32_wmma(const float* __restrict__ A, long sA,
                   const float* __restrict__ Bm, long sB,
                   float* __restrict__ C, long sC,
                   int M, int Nn, int K,
                   const float* __restrict__ bias,
                   const float* __restrict__ bscale, long sBs,
                   const float* __restrict__ rowscale)
{
    constexpr int BM = 128, BNT = 128, BK = 32;
    constexpr int LDA = BM + 4, LDB = BNT + 4;
    __shared__ float As[BK * LDA];   // k-major: As[k][m]
    __shared__ float Bs[BK * LDB];   // k-major: Bs[k][n]

    const int tid  = threadIdx.x;
    const int lane = tid & 31;
    const int wave = tid >> 5;          // 0..3
    const int wm   = wave >> 1;         // 0..1  (64-row stripes)
    const int wn   = wave & 1;          // 0..1  (64-col stripes)
    const int half = lane >> 4;         // ISA half-wave select
    const int lm   = lane & 15;

    const int bz = blockIdx.z;
    const int bm = blockIdx.y * BM;
    const int bn = blockIdx.x * BNT;

    const float* Ab = A  + (size_t)bz * sA;
    const float* Bb = Bm + (size_t)bz * sB;
    float*       Cb = C  + (size_t)bz * sC;
    const float* scb = BSCALE ? (bscale + (size_t)bz * sBs) : nullptr;

    v8f acc[4][4];
    const v8f vzero = {0.f,0.f,0.f,0.f,0.f,0.f,0.f,0.f};
#pragma unroll
    for (int mi = 0; mi < 4; ++mi)
#pragma unroll
        for (int ni = 0; ni < 4; ++ni) acc[mi][ni] = vzero;

    for (int kb = 0; kb < K; kb += BK) {
        // ---- stage A tile (128 x 32) into LDS, transposed to k-major ----
#pragma unroll
        for (int l = 0; l < 8; ++l) {
            int id  = tid + l * 128;     // 0..1023 float4 slots
            int row = id >> 3;           // 0..127  (m)
            int c4  = id & 7;            // 0..7    (k/4)
            const float4 v = *(const float4*)(Ab + (size_t)(bm + row) * K + kb + c4 * 4);
            As[(c4 * 4 + 0) * LDA + row] = v.x;
            As[(c4 * 4 + 1) * LDA + row] = v.y;
            As[(c4 * 4 + 2) * LDA + row] = v.z;
            As[(c4 * 4 + 3) * LDA + row] = v.w;
        }
        // ---- stage B tile ----
        if constexpr (!BT) {
            // B is [K x Nn] row-major: rows kb..kb+31, cols bn..bn+127
#pragma unroll
            for (int l = 0; l < 8; ++l) {
                int id  = tid + l * 128;
                int row = id >> 5;       // 0..31 (k)
                int c4  = id & 31;       // 0..31 (n/4)
                float4 v = *(const float4*)(Bb + (size_t)(kb + row) * Nn + bn + c4 * 4);
                if constexpr (BSCALE) {
                    float s = scb[kb + row];
                    v.x *= s; v.y *= s; v.z *= s; v.w *= s;
                }
                *(float4*)&Bs[row * LDB + c4 * 4] = v;
            }
        } else {
            // B given as Bt [Nn x K] row-major; transpose while staging
#pragma unroll
            for (int l = 0; l < 8; ++l) {
                int id  = tid + l * 128;
                int row = id >> 3;       // 0..127 (n)
                int c4  = id & 7;        // 0..7   (k/4)
                const float4 v = *(const float4*)(Bb + (size_t)(bn + row) * K + kb + c4 * 4);
                Bs[(c4 * 4 + 0) * LDB + row] = v.x;
                Bs[(c4 * 4 + 1) * LDB + row] = v.y;
                Bs[(c4 * 4 + 2) * LDB + row] = v.z;
                Bs[(c4 * 4 + 3) * LDB + row] = v.w;
            }
        }
        __syncthreads();

        // ---- compute: BK/4 = 8 k-steps, 16 WMMAs each ----
#pragma unroll
        for (int k0 = 0; k0 < BK; k0 += 4) {
            v2f afr[4], bfr[4];
#pragma unroll
            for (int mi = 0; mi < 4; ++mi) {
                int mcol = wm * 64 + mi * 16 + lm;
                afr[mi].x = As[(k0 + half * 2 + 0) * LDA + mcol];
                afr[mi].y = As[(k0 + half * 2 + 1) * LDA + mcol];
            }
#pragma unroll
            for (int ni = 0; ni < 4; ++ni) {
                int ncol = wn * 64 + ni * 16 + lm;
                bfr[ni].x = Bs[(k0 + half * 2 + 0) * LDB + ncol];
                bfr[ni].y = Bs[(k0 + half * 2 + 1) * LDB + ncol];
            }
#pragma unroll
            for (int mi = 0; mi < 4; ++mi)
#pragma unroll
                for (int ni = 0; ni < 4; ++ni)
                    acc[mi][ni] = __builtin_amdgcn_wmma_f32_16x16x4_f32(
                        false, afr[mi], false, bfr[ni],
                        (short)0, acc[mi][ni], false, false);
        }
        __syncthreads();
    }

    // ---- epilogue ----
#pragma unroll
    for (int mi = 0; mi < 4; ++mi) {
#pragma unroll
        for (int ni = 0; ni < 4; ++ni) {
            int col = bn + wn * 64 + ni * 16 + lm;
            float bv = 0.0f;
            if constexpr (BIAS) bv = bias[col];
#pragma unroll
            for (int r = 0; r < 8; ++r) {
                int row = bm + wm * 64 + mi * 16 + r + half * 8;
                float v = acc[mi][ni][r];
                if constexpr (BIAS) v += bv;
                if constexpr (RELU) v = v > 0.0f ? v : 0.0f;
                if constexpr (ROWSCALE) v *= rowscale[row];
                Cb[(size_t)row * Nn + col] = v;
            }
        }
    }
}

// =====================================================================
// Mean pool + attention logits.  h is already mask-scaled.
// out[b,hd] = sum_n h / (eps + masksum);  logits[b,n] = h.w_a + (mask-1)*NEG
// =====================================================================
__global__ __launch_bounds__(256)
void pool_logits_kernel(const float* __restrict__ h,
                        const float* __restrict__ mask,
                        const float* __restrict__ w_a,
                        float* __restrict__ out,
                        float* __restrict__ logits)
{
    __shared__ float red[256];
    const int b = blockIdx.x, tid = threadIdx.x;
    const float* hb = h + (size_t)b * NNODES * HDIM;
    const float* mb = mask + (size_t)b * NNODES;

    // masksum
    float ms = 0.0f;
    for (int n = tid; n < NNODES; n += 256) ms += mb[n];
    red[tid] = ms; __syncthreads();
    for (int s = 128; s > 0; s >>= 1) { if (tid < s) red[tid] += red[tid + s]; __syncthreads(); }
    const float masksum = red[0];
    __syncthreads();

    // column sums: 2 rows x 128 cols
    const int hd = tid & 127, r0 = tid >> 7;
    float csum = 0.0f;
    for (int n = r0; n < NNODES; n += 2) csum += hb[(size_t)n * HDIM + hd];
    red[r0 * 128 + hd] = csum; __syncthreads();
    if (tid < 128) out[(size_t)b * HDIM + tid] = (red[tid] + red[128 + tid]) / (EPSV + masksum);

    // logits: one row per wave, 4-wide dot per lane, wave32 xor-reduce
    const int wave = tid >> 5, lane = tid & 31;
    const float4 wv = *(const float4*)(w_a + lane * 4);
    for (int n = wave; n < NNODES; n += 8) {
        const float4 hv = *(const float4*)(hb + (size_t)n * HDIM + lane * 4);
        float d = hv.x * wv.x + hv.y * wv.y + hv.z * wv.z + hv.w * wv.w;
#pragma unroll
        for (int off = 16; off > 0; off >>= 1) d += __shfl_xor(d, off, 32);
        if (lane == 0) logits[(size_t)b * NNODES + n] = d + (mb[n] - 1.0f) * NEGV;
    }
}

// =====================================================================
// Per-batch softmax over N + full bitonic top-k (descending, index tie
// break ascending to match lax.top_k). Writes att, top_idx, k_per, new_mask.
// =====================================================================
__global__ __launch_bounds__(1024)
void softmax_topk_kernel(const float* __restrict__ logits,
                         const float* __restrict__ mask,
                         float* __restrict__ att,
                         int* __restrict__ top_idx,
                         int* __restrict__ k_per,
                         float* __restrict__ newmask)
{
    __shared__ float skey[NNODES];
    __shared__ int   sidx[NNODES];
    __shared__ float red[1024];
    const int b = blockIdx.x, tid = threadIdx.x;
    const float* lb = logits + (size_t)b * NNODES;
    const float* mb = mask + (size_t)b * NNODES;

    const float v0 = lb[tid], v1 = lb[tid + 1024];
    red[tid] = fmaxf(v0, v1); __syncthreads();
    for (int s = 512; s > 0; s >>= 1) { if (tid < s) red[tid] = fmaxf(red[tid], red[tid + s]); __syncthreads(); }
    const float mx = red[0]; __syncthreads();

    const float e0 = expf(v0 - mx), e1 = expf(v1 - mx);
    red[tid] = e0 + e1; __syncthreads();
    for (int s = 512; s > 0; s >>= 1) { if (tid < s) red[tid] += red[tid + s]; __syncthreads(); }
    const float denom = red[0]; __syncthreads();

    const float a0 = e0 / denom, a1 = e1 / denom;
    att[(size_t)b * NNODES + tid]        = a0;
    att[(size_t)b * NNODES + tid + 1024] = a1;
    skey[tid] = a0; skey[tid + 1024] = a1;
    sidx[tid] = tid; sidx[tid + 1024] = tid + 1024;

    // masksum -> k_per
    red[tid] = mb[tid] + mb[tid + 1024]; __syncthreads();
    for (int s = 512; s > 0; s >>= 1) { if (tid < s) red[tid] += red[tid + s]; __syncthreads(); }
    const int kp = (int)ceilf(PERCENTV * red[0]);

    // bitonic sort, descending by key, ascending index on ties
    for (int size = 2; size <= NNODES; size <<= 1) {
        for (int stride = size >> 1; stride > 0; stride >>= 1) {
            __syncthreads();
            const int i = 2 * tid - (tid & (stride - 1));
            const int j = i + stride;
            const bool desc = ((i & size) == 0);
            const float ki = skey[i], kj = skey[j];
            const int   ii = sidx[i], ij = sidx[j];
            const bool inorder = (ki > kj) || (ki == kj && ii < ij);
            if (inorder != desc) {
                skey[i] = kj; skey[j] = ki;
                sidx[i] = ij; sidx[j] = ii;
            }
        }
    }
    __syncthreads();

    if (tid < KMAXV) {
        top_idx[(size_t)b * KMAXV + tid] = sidx[tid];
        newmask[(size_t)b * KMAXV + tid] = (tid < kp) ? 1.0f : 0.0f;
        if (tid == 0) k_per[b] = kp;
    }
}

// gather selected adjacency rows * valid
__global__ __launch_bounds__(256)
void gather_assign_kernel(const float* __restrict__ adj,
                          const int* __restrict__ top_idx,
                          const int* __restrict__ k_per,
                          float* __restrict__ assign)
{
    const int rowg = blockIdx.x;           // 0..B*KMAX-1
    const int b = rowg >> 9, k = rowg & (KMAXV - 1);
    const int src = top_idx[rowg];
    const float valid = (k < k_per[b]) ? 1.0f : 0.0f;
    const float* arow = adj + ((size_t)b * NNODES + src) * NNODES;
    float* drow = assign + (size_t)rowg * NNODES;
    for (int n = threadIdx.x * 4; n < NNODES; n += 256 * 4) {
        float4 v = *(const float4*)(arow + n);
        v.x *= valid; v.y *= valid; v.z *= valid; v.w *= valid;
        *(float4*)(drow + n) = v;
    }
}

// column sums over k (assign is [B, KMAX, N])
__global__ __launch_bounds__(256)
void colsum_kernel(const float* __restrict__ assign, float* __restrict__ colsumb)
{
    const int g = blockIdx.x * 256 + threadIdx.x;   // 0..B*N-1
    const int b = g >> 11, n = g & (NNODES - 1);
    const float* ab = assign + (size_t)b * KMAXV * NNODES + n;
    float s = 0.0f;
    for (int k = 0; k < KMAXV; ++k) s += ab[(size_t)k * NNODES];
    colsumb[g] = s;
}

// assign /= (colsum + eps)
__global__ __launch_bounds__(256)
void norm_assign_kernel(float* __restrict__ assign, const float* __restrict__ colsumb)
{
    const size_t g = ((size_t)blockIdx.x * 256 + threadIdx.x) * 4;
    const int n = (int)(g & (NNODES - 1));
    const int b = (int)(g >> 20);                   // KMAX*N = 2^20
    float4 v = *(const float4*)(assign + g);
    const float* cs = colsumb + b * NNODES + n;
    v.x /= (cs[0] + EPSV); v.y /= (cs[1] + EPSV);
    v.z /= (cs[2] + EPSV); v.w /= (cs[3] + EPSV);
    *(float4*)(assign + g) = v;
}

// =====================================================================
extern "C" void kernel_launch(void* const* d_in, const int* in_sizes, int n_in,
                              void* d_out, int out_size, void* d_ws, size_t ws_size,
                              hipStream_t stream)
{
    (void)in_sizes; (void)n_in; (void)out_size; (void)ws_size;
    const float* X    = (const float*)d_in[0];
    const float* adj  = (const float*)d_in[1];
    const float* mask = (const float*)d_in[2];
    const float* W1   = (const float*)d_in[3];
    const float* b1   = (const float*)d_in[4];
    const float* W2   = (const float*)d_in[5];
    const float* b2   = (const float*)d_in[6];
    const float* w_a  = (const float*)d_in[7];
    float* out = (float*)d_out;

    // output layout: out[B,HD] | H_out[B,KMAX,HD] | new_adj[B,KMAX,KMAX] | new_mask[B,KMAX]
    const size_t HOUT_OFF    = (size_t)BATCH * HDIM;                       // 2048
    const size_t NEWADJ_OFF  = HOUT_OFF + (size_t)BATCH * KMAXV * HDIM;    // 1050624
    const size_t NEWMASK_OFF = NEWADJ_OFF + (size_t)BATCH * KMAXV * KMAXV; // 5244928

    // workspace layout
    float* ws      = (float*)d_ws;
    float* bufA    = ws;                                           // B*N*HD
    float* bufB    = bufA   + (size_t)BATCH * NNODES * HDIM;       // B*N*HD
    float* attb    = bufB   + (size_t)BATCH * NNODES * HDIM;       // B*N (logits -> att)
    float* colsumb = attb   + (size_t)BATCH * NNODES;              // B*N
    float* assign  = colsumb+ (size_t)BATCH * NNODES;              // B*KMAX*N
    float* Tbuf    = assign + (size_t)BATCH * KMAXV * NNODES;      // B*KMAX*N
    int*   top_idx = (int*)(Tbuf + (size_t)BATCH * KMAXV * NNODES);
    int*   kper    = top_idx + BATCH * KMAXV;

    const long NN = (long)NNODES * NNODES;

    // 1) h1 = relu(adj @ X)  [B: M=2048,N=128,K=2048]
    gemm_f32_wmma<true,false,false,false,false>
        <<<dim3(DDIM/128, NNODES/128, BATCH), 128, 0, stream>>>(
            adj, NN, X, (long)NNODES*DDIM, bufA, (long)NNODES*HDIM,
            NNODES, DDIM, NNODES, nullptr, nullptr, 0, nullptr);

    // 2) h2 = h1 @ W1 + b1   [flat: M=B*N, N=128, K=128]
    gemm_f32_wmma<false,true,false,false,false>
        <<<dim3(HDIM/128, (BATCH*NNODES)/128, 1), 128, 0, stream>>>(
            bufA, 0, W1, 0, bufB, 0,
            BATCH*NNODES, HDIM, DDIM, b1, nullptr, 0, nullptr);

    // 3) t = adj @ h2
    gemm_f32_wmma<false,false,false,false,false>
        <<<dim3(HDIM/128, NNODES/128, BATCH), 128, 0, stream>>>(
            adj, NN, bufB, (long)NNODES*HDIM, bufA, (long)NNODES*HDIM,
            NNODES, HDIM, NNODES, nullptr, nullptr, 0, nullptr);

    // 4) h = mask * (t @ W2 + b2)   [flat, row scale by mask]
    gemm_f32_wmma<false,true,false,false,true>
        <<<dim3(HDIM/128, (BATCH*NNODES)/128, 1), 128, 0, stream>>>(
            bufA, 0, W2, 0, bufB, 0,
            BATCH*NNODES, HDIM, HDIM, b2, nullptr, 0, mask);

    // 5) mean pool -> out ; attention logits
    pool_logits_kernel<<<BATCH, 256, 0, stream>>>(bufB, mask, w_a, out, attb);

    // 6) softmax + top-k -> att, top_idx, k_per, new_mask
    softmax_topk_kernel<<<BATCH, 1024, 0, stream>>>(
        attb, mask, attb, top_idx, kper, out + NEWMASK_OFF);

    // 7) gather adjacency rows, 8) column sums, 9) normalize
    gather_assign_kernel<<<BATCH*KMAXV, 256, 0, stream>>>(adj, top_idx, kper, assign);
    colsum_kernel<<<(BATCH*NNODES)/256, 256, 0, stream>>>(assign, colsumb);
    norm_assign_kernel<<<(BATCH*KMAXV*NNODES/4)/256, 256, 0, stream>>>(assign, colsumb);

    // 10) H_out = assign @ (att ⊙ h)   (att folded as B-row scale)
    gemm_f32_wmma<false,false,false,true,false>
        <<<dim3(HDIM/128, KMAXV/128, BATCH), 128, 0, stream>>>(
            assign, (long)KMAXV*NNODES, bufB, (long)NNODES*HDIM,
            out + HOUT_OFF, (long)KMAXV*HDIM,
            KMAXV, HDIM, NNODES, nullptr, attb, NNODES, nullptr);

    // 11) T = assign @ adj
    gemm_f32_wmma<false,false,false,false,false>
        <<<dim3(NNODES/128, KMAXV/128, BATCH), 128, 0, stream>>>(
            assign, (long)KMAXV*NNODES, adj, NN, Tbuf, (long)KMAXV*NNODES,
            KMAXV, NNODES, NNODES, nullptr, nullptr, 0, nullptr);

    // 12) new_adj = T @ assign^T   (B transposed)
    gemm_f32_wmma<false,false,true,false,false>
        <<<dim3(KMAXV/128, KMAXV/128, BATCH), 128, 0, stream>>>(
            Tbuf, (long)KMAXV*NNODES, assign, (long)KMAXV*NNODES,
            out + NEWADJ_OFF, (long)KMAXV*KMAXV,
            KMAXV, KMAXV, NNODES, nullptr, nullptr, 0, nullptr);
}